// Q_Conv2d_39427799777378
// MI455X (gfx1250) — compile-verified
//
#include <hip/hip_runtime.h>

typedef __attribute__((ext_vector_type(2))) float v2f;
typedef __attribute__((ext_vector_type(8))) float v8f;

namespace {

constexpr int kWaves = 4;         // 128-thread blocks = 4 wave32
constexpr int kLdsPerWave = 512;  // Sr, Si : 2 x 256 floats

// Net inverse permutation of the CNOT ring CNOT(0,1)..CNOT(7,0).
// Amplitude index bit (7-q) <-> qubit q (axis-0 of the jnp state is the MSB).
__device__ __forceinline__ int cnot_ring_invperm(int j) {
#pragma unroll
  for (int q = 0; q < 8; ++q) {
    const int cbit = 7 - q;
    const int tbit = 7 - ((q + 1) & 7);
    j ^= ((j >> cbit) & 1) << tbit;
  }
  return j;
}

// A-fragment of a row-major 16x16 f32 matrix, k-block kblk (A[:,4k:4k+4]).
// ISA 7.12.2: lanes 0-15 M=0..15 K={0,1}, lanes 16-31 M=0..15 K={2,3}.
__device__ __forceinline__ v2f load_A16(const float* buf, int kblk, int lane) {
  const int m = lane & 15;
  const int kb = (lane >> 4) << 1;
  v2f a;
  a.x = buf[m * 16 + kblk * 4 + kb + 0];
  a.y = buf[m * 16 + kblk * 4 + kb + 1];
  return a;
}

// B-fragment (B[4k:4k+4, :]): row K striped across lanes within a VGPR.
__device__ __forceinline__ v2f load_B16(const float* buf, int kblk, int lane) {
  const int n = lane & 15;
  const int kb = (lane >> 4) << 1;
  v2f b;
  b.x = buf[(kblk * 4 + kb + 0) * 16 + n];
  b.y = buf[(kblk * 4 + kb + 1) * 16 + n];
  return b;
}

// acc += A(16x16) * B(16x16), chained over 4 K-blocks of v_wmma_f32_16x16x4_f32.
__device__ __forceinline__ v8f mm16_acc(const float* Abuf, const float* Bbuf,
                                        v8f acc, int lane) {
#pragma unroll
  for (int k = 0; k < 4; ++k) {
    v2f a = load_A16(Abuf, k, lane);
    v2f b = load_B16(Bbuf, k, lane);
    acc = __builtin_amdgcn_wmma_f32_16x16x4_f32(false, a, false, b, (short)0,
                                                acc, false, false);
  }
  return acc;
}

// Build 16x16 Kron gate RX(a0)xRX(a1)xRX(a2)xRX(a3): entry(r,c) has magnitude
// prod(cos/sin by differing bits) and phase (-i)^popcount(r^c).
__device__ __forceinline__ void build_gate(float* Gr, float* Gi, float* Gin,
                                           const float* hc, const float* hs,
                                           int lane) {
#pragma unroll
  for (int j = 0; j < 8; ++j) {
    const int e = lane * 8 + j;
    const int d = (e >> 4) ^ (e & 15);
    float prod = 1.0f;
#pragma unroll
    for (int k = 0; k < 4; ++k) {
      prod *= ((d >> k) & 1) ? hs[3 - k] : hc[3 - k];  // bit k <-> qubit 3-k
    }
    const int dp = __builtin_popcount((unsigned)d) & 3;
    const float re = (dp == 0) ? prod : ((dp == 2) ? -prod : 0.0f);
    const float im = (dp == 1) ? -prod : ((dp == 3) ? prod : 0.0f);
    Gr[e] = re;
    Gi[e] = im;
    Gin[e] = -im;
  }
}

// Store C/D fragment to row-major LDS, optionally applying the CNOT-ring
// permutation at store time (new[i] = old[perm(i)]  =>  dst = invperm(src)).
__device__ __forceinline__ void store_state(float* Sr, float* Si, v8f sr,
                                            v8f si, int lane, bool cnot) {
  const int n = lane & 15;
  const int mb = (lane >> 4) << 3;
#pragma unroll
  for (int v = 0; v < 8; ++v) {
    const int j = (mb + v) * 16 + n;
    const int d = cnot ? cnot_ring_invperm(j) : j;
    Sr[d] = sr[v];
    Si[d] = si[v];
  }
}

}  // namespace

// ---------------------------------------------------------------------------
// Kernel 1: precompute the 16 (channel,layer) gate-sets x 2 sides into d_ws.
// Layout: gates[g * 768], g = (ch*4 + layer)*2 + side; planes Gr/Gi/-Gi @ 256.
// 32 gate-sets -> 96 KB, L2-resident for the main kernel.
// ---------------------------------------------------------------------------
__global__ void __launch_bounds__(128)
qconv2d_build_gates_kernel(const float* __restrict__ wts,
                           float* __restrict__ gates) {
  const int lane = threadIdx.x & 31;
  const int wv = blockIdx.x * kWaves + (threadIdx.x >> 5);  // 0..31
  const int set = wv >> 1;   // ch*4 + layer
  const int side = wv & 1;   // 0 = hi (qubits 0..3), 1 = lo (qubits 4..7)
  float hc[4], hs[4];
#pragma unroll
  for (int q = 0; q < 4; ++q) {
    float s, c;
    sincosf(wts[set * 8 + side * 4 + q] * 0.5f, &s, &c);
    hc[q] = c;
    hs[q] = s;
  }
  float* base = gates + wv * 768;
  build_gate(base, base + 256, base + 512, hc, hs, lane);
}

// ---------------------------------------------------------------------------
// Kernel 2: one wave = one circuit. Embed via the rank-1 outer-product
// identity, then 4 layers of complex 16x16 gate multiplies on the f32 WMMA
// pipe, CNOT ring folded into the LDS store permutation.
// ---------------------------------------------------------------------------
__global__ void __launch_bounds__(kWaves * 32)
qconv2d_wmma_kernel(const float* __restrict__ x, const float* __restrict__ gates,
                    float* __restrict__ out) {
  __shared__ float lds[kWaves * kLdsPerWave];
  const int lane = threadIdx.x & 31;
  const int wave = threadIdx.x >> 5;
  const int gid = blockIdx.x * kWaves + wave;  // [0, 32768), all blocks full

  const int ch = gid >> 13;        // output channel 0..3
  const int p = gid & 8191;        // patch id
  const int b = p >> 12;           // batch 0..1
  const int ph = (p >> 6) & 63;
  const int pw = p & 63;

  float* Sr = lds + wave * kLdsPerWave;  // wave-private: no barriers needed
  float* Si = Sr + 256;

  // ---- Embed: state = RX(angles) |0..0> is rank-1; amplitude i is
  // (-i)^popcount(i) * prod_k (bit_k(i) ? sin : cos) of qubit (7-k).
  float hc[8], hs[8];
#pragma unroll
  for (int q = 0; q < 8; ++q) {  // q = c*4 + kh*2 + kw (patch order)
    const int c = q >> 2, kh = (q >> 1) & 1, kw = q & 1;
    float s, cth;
    sincosf(x[((b * 2 + c) * 128 + (2 * ph + kh)) * 128 + (2 * pw + kw)] * 0.5f,
            &s, &cth);
    hc[q] = cth;
    hs[q] = s;
  }
#pragma unroll
  for (int j = 0; j < 8; ++j) {
    const int i = lane * 8 + j;
    float prod = 1.0f;
#pragma unroll
    for (int k = 0; k < 8; ++k) {
      prod *= ((i >> k) & 1) ? hs[7 - k] : hc[7 - k];
    }
    const int dp = __builtin_popcount((unsigned)i) & 3;
    Sr[i] = (dp == 0) ? prod : ((dp == 2) ? -prod : 0.0f);
    Si[i] = (dp == 1) ? -prod : ((dp == 3) ? prod : 0.0f);
  }

  const v8f zero = {0.f, 0.f, 0.f, 0.f, 0.f, 0.f, 0.f, 0.f};

  // ---- Layers: S' = Ghi * S * Glo (complex), then CNOT-ring permutation.
  for (int l = 0; l < 4; ++l) {
    const float* GA = gates + (((ch * 4 + l) * 2 + 0) * 768);  // hi: Gr,Gi,-Gi
    const float* GB = gates + (((ch * 4 + l) * 2 + 1) * 768);  // lo: Gr,Gi,-Gi

    // Left multiply (rows = qubits 0..3):  S' = G * S
    v8f srN = mm16_acc(GA, Sr, zero, lane);
    srN = mm16_acc(GA + 512, Si, srN, lane);  // - Gi*Si
    v8f siN = mm16_acc(GA, Si, zero, lane);
    siN = mm16_acc(GA + 256, Sr, siN, lane);
    store_state(Sr, Si, srN, siN, lane, false);

    // Right multiply (cols = qubits 4..7):  S' = S * H  (H symmetric)
    srN = mm16_acc(Sr, GB, zero, lane);
    srN = mm16_acc(Si, GB + 512, srN, lane);  // - Si*Hi
    siN = mm16_acc(Sr, GB + 256, zero, lane);
    siN = mm16_acc(Si, GB, siN, lane);
    store_state(Sr, Si, srN, siN, lane, /*cnot=*/true);
  }

  // ---- <Z> on qubit 7 = sum |s_i|^2 * (1 - 2*(i&1)); i&1 is column parity.
  const int n = lane & 15;
  const int mb = (lane >> 4) << 3;
  float acc = 0.0f;
#pragma unroll
  for (int v = 0; v < 8; ++v) {
    const int j = (mb + v) * 16 + n;
    const float a = Sr[j];
    const float bb = Si[j];
    acc += a * a + bb * bb;
  }
  acc = (n & 1) ? -acc : acc;
#pragma unroll
  for (int off = 16; off > 0; off >>= 1) acc += __shfl_xor(acc, off, 32);
  if (lane == 0) out[((b * 4 + ch) * 64 + ph) * 64 + pw] = acc;
}

extern "C" void kernel_launch(void* const* d_in, const int* in_sizes, int n_in,
                              void* d_out, int out_size, void* d_ws,
                              size_t ws_size, hipStream_t stream) {
  (void)in_sizes; (void)n_in; (void)out_size; (void)ws_size;
  const float* x = (const float*)d_in[0];  // (2,2,128,128) f32
  const float* w = (const float*)d_in[1];  // (4,4,8) f32
  float* out = (float*)d_out;              // (2,4,64,64) f32
  float* gates = (float*)d_ws;             // 32 * 768 floats = 96 KB

  // Precompute (channel,layer) gate matrices once (32 waves).
  qconv2d_build_gates_kernel<<<8, kWaves * 32, 0, stream>>>(w, gates);

  // 32768 circuits, one wave each.
  const int n_circuits = 4 * 2 * 64 * 64;
  qconv2d_wmma_kernel<<<n_circuits / kWaves, kWaves * 32, 0, stream>>>(x, gates,
                                                                      out);
}